// EdgeAttentionLayer_75788992905486
// MI455X (gfx1250) — compile-verified
//
#include <hip/hip_runtime.h>

// CDNA5 / gfx1250: wave32, f32 WMMA 16x16x4 (A,B = 2 VGPRs; C/D = 8 VGPRs)
typedef float v2f __attribute__((ext_vector_type(2)));
typedef float v8f __attribute__((ext_vector_type(8)));

#define SLOPE   0.2f
#define NEG_INF (-3.402823466e38f)

__device__ __forceinline__ float lrelu(float x) { return x > 0.f ? x : SLOPE * x; }

__device__ __forceinline__ v8f wmma_f32(v2f a, v2f b, v8f c) {
  // D = A(16x4) * B(4x16) + C(16x16), fp32 throughout
  return __builtin_amdgcn_wmma_f32_16x16x4_f32(false, a, false, b, (short)0, c, false, false);
}

// ---------------------------------------------------------------------------
// node_feat[n] = node_emb[n] @ W_node + b_node   (N x 64, K=64)
// ---------------------------------------------------------------------------
__global__ void __launch_bounds__(256) k_node_feat(
    const float* __restrict__ node_emb, const float* __restrict__ W_node,
    const float* __restrict__ b_node, float* __restrict__ node_feat, int ntiles) {
  int wave = threadIdx.x >> 5, lane = threadIdx.x & 31;
  int tile = blockIdx.x * 8 + wave;
  if (tile >= ntiles) return;                       // wave-uniform: WMMA EXEC stays all-1s
  int half = lane >> 4, mrow = lane & 15;
  int n0 = tile * 16;
  const float* arow = node_emb + (size_t)(n0 + mrow) * 64;
  v8f acc[4] = {};
  for (int k0 = 0; k0 < 64; k0 += 4) {
    int k = k0 + half * 2;
    v2f a; a.x = arow[k]; a.y = arow[k + 1];
#pragma unroll
    for (int nt = 0; nt < 4; ++nt) {
      int n = nt * 16 + mrow;
      v2f b; b.x = W_node[k * 64 + n]; b.y = W_node[(k + 1) * 64 + n];
      acc[nt] = wmma_f32(a, b, acc[nt]);
    }
  }
#pragma unroll
  for (int nt = 0; nt < 4; ++nt) {
    int n = nt * 16 + mrow;
    float bn = b_node[n];
#pragma unroll
    for (int r = 0; r < 8; ++r)
      node_feat[(size_t)(n0 + r + 8 * half) * 64 + n] = acc[nt][r] + bn;
  }
}

// ---------------------------------------------------------------------------
// Main edge kernel: one wave handles a 16-edge tile.
// Produces updated[E,64], logits[E,64], per-block channel maxes.
// ---------------------------------------------------------------------------
__global__ void __launch_bounds__(256) k_edge_main(
    const float* __restrict__ edge_emb, const float* __restrict__ node_emb,
    const float* __restrict__ W_attn, const float* __restrict__ b_attn,
    const float* __restrict__ W_upd,  const float* __restrict__ b_upd,
    const float* __restrict__ W_edge, const float* __restrict__ b_edge,
    const float* __restrict__ node_feat, const long long* __restrict__ edge_index,
    float* __restrict__ updated, float* __restrict__ logits,
    float* __restrict__ partialMax, long long E, int ntiles) {
  __shared__ float upd_s[8][16][68];   // stride 68 -> bank 4*row: conflict-free A reads
  __shared__ float redMax[8][64];
  int wave = threadIdx.x >> 5, lane = threadIdx.x & 31;
  int half = lane >> 4, mrow = lane & 15;
  int tile = blockIdx.x * 8 + wave;
  float mx[4] = {NEG_INF, NEG_INF, NEG_INF, NEG_INF};

  if (tile < ntiles) {                              // wave-uniform
    long long e0 = (long long)tile * 16;
    long long e  = e0 + mrow;
    int sI = (int)edge_index[e];
    int dI = (int)edge_index[E + e];
    const float* srcRow = node_emb + (size_t)sI * 64;   // L2-resident (25.6 MB)
    const float* dstRow = node_emb + (size_t)dI * 64;
    const float* embRow = edge_emb + (size_t)e * 64;    // streamed, tile-contiguous
    __builtin_prefetch(embRow + 16 * 64, 0, 1);         // next tile -> global_prefetch_b8

    // ---- gate = sigmoid(lrelu(combined . W_attn + b_attn)), split across halves ----
    float g = 0.f;
    {
      int kb = half * 96;
#pragma unroll 8
      for (int k = kb; k < kb + 96; ++k) {
        float v = (k < 64) ? srcRow[k] : (k < 128 ? dstRow[k - 64] : embRow[k - 128]);
        g += v * W_attn[k];
      }
    }
    g += __shfl_xor(g, 16, 32);
    float gate = 1.f / (1.f + __expf(-lrelu(g + b_attn[0])));
    float gv[8];
#pragma unroll
    for (int r = 0; r < 8; ++r) gv[r] = __shfl(gate, r + 8 * half, 32);

    // ---- updated = (combined @ W_upd + b_upd) * gate : M=16, N=64, K=192 ----
    v8f acc[4] = {};
    for (int k0 = 0; k0 < 192; k0 += 4) {
      int k = k0 + half * 2;
      const float* ap; int kc;
      if (k < 64)       { ap = srcRow; kc = k; }
      else if (k < 128) { ap = dstRow; kc = k - 64; }
      else              { ap = embRow; kc = k - 128; }
      v2f a; a.x = ap[kc]; a.y = ap[kc + 1];
#pragma unroll
      for (int nt = 0; nt < 4; ++nt) {
        int n = nt * 16 + mrow;
        v2f b; b.x = W_upd[k * 64 + n]; b.y = W_upd[(k + 1) * 64 + n];
        acc[nt] = wmma_f32(a, b, acc[nt]);
      }
    }
#pragma unroll
    for (int nt = 0; nt < 4; ++nt) {
      int n = nt * 16 + mrow;
      float bu = b_upd[n];
#pragma unroll
      for (int r = 0; r < 8; ++r)
        upd_s[wave][r + 8 * half][n] = (acc[nt][r] + bu) * gv[r];
    }
    // coalesced float4 store of the updated tile (16x64) from LDS
#pragma unroll
    for (int i = 0; i < 8; ++i) {
      int f = i * 128 + lane * 4, row = f >> 6, col = f & 63;
      float4 v = *(const float4*)&upd_s[wave][row][col];
      *(float4*)&updated[(size_t)(e0 + row) * 64 + col] = v;
    }

    // ---- logits = lrelu(updated @ W_edge + b_edge + nf[src] + nf[dst]) ----
    v8f acc2[4] = {};
    for (int k0 = 0; k0 < 64; k0 += 4) {
      int k = k0 + half * 2;
      v2f a; a.x = upd_s[wave][mrow][k]; a.y = upd_s[wave][mrow][k + 1];
#pragma unroll
      for (int nt = 0; nt < 4; ++nt) {
        int n = nt * 16 + mrow;
        v2f b; b.x = W_edge[k * 64 + n]; b.y = W_edge[(k + 1) * 64 + n];
        acc2[nt] = wmma_f32(a, b, acc2[nt]);
      }
    }
    int sL[8], dL[8];
#pragma unroll
    for (int r = 0; r < 8; ++r) {
      sL[r] = __shfl(sI, r + 8 * half, 32);
      dL[r] = __shfl(dI, r + 8 * half, 32);
    }
#pragma unroll
    for (int nt = 0; nt < 4; ++nt) {
      int n = nt * 16 + mrow;
      float be = b_edge[n];
#pragma unroll
      for (int r = 0; r < 8; ++r) {
        float l = acc2[nt][r] + be + node_feat[(size_t)sL[r] * 64 + n]
                                   + node_feat[(size_t)dL[r] * 64 + n];
        l = lrelu(l);
        mx[nt] = fmaxf(mx[nt], l);
        upd_s[wave][r + 8 * half][n] = l;           // reuse tile buffer for logits
      }
    }
#pragma unroll
    for (int i = 0; i < 8; ++i) {
      int f = i * 128 + lane * 4, row = f >> 6, col = f & 63;
      float4 v = *(const float4*)&upd_s[wave][row][col];
      *(float4*)&logits[(size_t)(e0 + row) * 64 + col] = v;
    }
#pragma unroll
    for (int nt = 0; nt < 4; ++nt)
      mx[nt] = fmaxf(mx[nt], __shfl_xor(mx[nt], 16, 32));
  }

  // block-level per-channel max (deterministic, no atomics)
  if (lane < 16) {
#pragma unroll
    for (int nt = 0; nt < 4; ++nt) redMax[wave][nt * 16 + lane] = mx[nt];
  }
  __syncthreads();
  if (threadIdx.x < 64) {
    float m = redMax[0][threadIdx.x];
#pragma unroll
    for (int w2 = 1; w2 < 8; ++w2) m = fmaxf(m, redMax[w2][threadIdx.x]);
    partialMax[(size_t)blockIdx.x * 64 + threadIdx.x] = m;
  }
}

// ---------------------------------------------------------------------------
// Column reduction of [rows,64] partials -> [64] (isMax ? max : sum)
// ---------------------------------------------------------------------------
__global__ void k_colreduce(const float* __restrict__ part, int rows,
                            float* __restrict__ out, int isMax) {
  __shared__ float red[4][64];
  int c = threadIdx.x & 63, g = threadIdx.x >> 6;
  float a = isMax ? NEG_INF : 0.f;
  for (int r = g; r < rows; r += 4) {
    float v = part[(size_t)r * 64 + c];
    a = isMax ? fmaxf(a, v) : (a + v);
  }
  red[g][c] = a;
  __syncthreads();
  if (threadIdx.x < 64) {
    float m = red[0][threadIdx.x];
    for (int i = 1; i < 4; ++i)
      m = isMax ? fmaxf(m, red[i][threadIdx.x]) : (m + red[i][threadIdx.x]);
    out[threadIdx.x] = m;
  }
}

// ---------------------------------------------------------------------------
// Per-channel sum of exp(logit - max): per-block partials (coalesced float4)
// ---------------------------------------------------------------------------
__global__ void __launch_bounds__(256) k_expsum(
    const float* __restrict__ logits, const float* __restrict__ chMax,
    float* __restrict__ partialSum, long long E) {
  __shared__ float red[16][16][4];
  int tid = threadIdx.x, c4 = tid & 15, rg = tid >> 4;
  float4 mx = ((const float4*)chMax)[c4];
  float4 s = make_float4(0.f, 0.f, 0.f, 0.f);
  long long r = (long long)blockIdx.x * 16 + rg;
  long long rstride = (long long)gridDim.x * 16;
  for (; r < E; r += rstride) {
    float4 v = ((const float4*)(logits + r * 64))[c4];
    s.x += __expf(v.x - mx.x); s.y += __expf(v.y - mx.y);
    s.z += __expf(v.z - mx.z); s.w += __expf(v.w - mx.w);
  }
  red[c4][rg][0] = s.x; red[c4][rg][1] = s.y; red[c4][rg][2] = s.z; red[c4][rg][3] = s.w;
  __syncthreads();
  if (tid < 64) {
    int cc = tid >> 2, j = tid & 3;
    float t = 0.f;
#pragma unroll
    for (int i = 0; i < 16; ++i) t += red[cc][i][j];
    partialSum[(size_t)blockIdx.x * 64 + tid] = t;
  }
}

// ---------------------------------------------------------------------------
// out = updated * exp(logit - max) / sum  (streaming float4)
// ---------------------------------------------------------------------------
__global__ void __launch_bounds__(256) k_final(
    const float* __restrict__ updated, const float* __restrict__ logits,
    const float* __restrict__ chMax, const float* __restrict__ chSum,
    float* __restrict__ out, long long nvec) {
  long long i = (long long)blockIdx.x * blockDim.x + threadIdx.x;
  long long stride = (long long)gridDim.x * blockDim.x;
  for (; i < nvec; i += stride) {
    int c4 = (int)(i & 15);
    float4 mx = ((const float4*)chMax)[c4];
    float4 sm = ((const float4*)chSum)[c4];
    float4 u = ((const float4*)updated)[i];
    float4 l = ((const float4*)logits)[i];
    float4 o;
    o.x = u.x * __expf(l.x - mx.x) / sm.x;
    o.y = u.y * __expf(l.y - mx.y) / sm.y;
    o.z = u.z * __expf(l.z - mx.z) / sm.z;
    o.w = u.w * __expf(l.w - mx.w) / sm.w;
    ((float4*)out)[i] = o;
  }
}

// ---------------------------------------------------------------------------
extern "C" void kernel_launch(void* const* d_in, const int* in_sizes, int n_in,
                              void* d_out, int out_size, void* d_ws, size_t ws_size,
                              hipStream_t stream) {
  const float* edge_emb = (const float*)d_in[0];
  // d_in[1] = edge_attr (unused by the reference)
  const float* node_emb = (const float*)d_in[2];
  const float* W_attn   = (const float*)d_in[3];
  const float* b_attn   = (const float*)d_in[4];
  const float* W_upd    = (const float*)d_in[5];
  const float* b_upd    = (const float*)d_in[6];
  const float* W_edge   = (const float*)d_in[7];
  const float* b_edge   = (const float*)d_in[8];
  const float* W_node   = (const float*)d_in[9];
  const float* b_node   = (const float*)d_in[10];
  const long long* edge_index = (const long long*)d_in[11];  // int64 [2,E]

  long long E = in_sizes[0] / 64;
  long long N = in_sizes[2] / 64;
  float* out = (float*)d_out;
  (void)n_in; (void)out_size; (void)ws_size;

  char* ws = (char*)d_ws;
  size_t off = 0;
  auto take = [&](size_t bytes) {
    char* p = ws + off;
    off = (off + bytes + 255) & ~(size_t)255;
    return p;
  };
  float* node_feat = (float*)take((size_t)N * 64 * 4);
  float* updated   = (float*)take((size_t)E * 64 * 4);
  float* logits    = (float*)take((size_t)E * 64 * 4);

  int ntilesE = (int)(E / 16);            // E = 1,000,000 -> 62,500 tiles
  int blocksE = (ntilesE + 7) / 8;
  int ntilesN = (int)(N / 16);            // N = 100,000 -> 6,250 tiles
  int blocksN = (ntilesN + 7) / 8;
  const int SUMBLOCKS = 4096;
  float* partialMax = (float*)take((size_t)blocksE * 64 * 4);
  float* partialSum = (float*)take((size_t)SUMBLOCKS * 64 * 4);
  float* chMax      = (float*)take(64 * 4);
  float* chSum      = (float*)take(64 * 4);

  k_node_feat<<<blocksN, 256, 0, stream>>>(node_emb, W_node, b_node, node_feat, ntilesN);
  k_edge_main<<<blocksE, 256, 0, stream>>>(edge_emb, node_emb, W_attn, b_attn,
                                           W_upd, b_upd, W_edge, b_edge, node_feat,
                                           edge_index, updated, logits, partialMax,
                                           E, ntilesE);
  k_colreduce<<<1, 256, 0, stream>>>(partialMax, blocksE, chMax, 1);
  k_expsum<<<SUMBLOCKS, 256, 0, stream>>>(logits, chMax, partialSum, E);
  k_colreduce<<<1, 256, 0, stream>>>(partialSum, SUMBLOCKS, chSum, 0);
  k_final<<<4096, 256, 0, stream>>>(updated, logits, chMax, chSum, out, E * 16);
}